// GCN_22385369547130
// MI455X (gfx1250) — compile-verified
//
#include <hip/hip_runtime.h>
#include <math.h>

typedef __attribute__((ext_vector_type(2))) float v2f;
typedef __attribute__((ext_vector_type(8))) float v8f;

#define HID 64
#define TPB 256

// ---------- small utility kernels ----------
__global__ void k_fill(float* p, float v, int n) {
    int i = blockIdx.x * blockDim.x + threadIdx.x;
    if (i < n) p[i] = v;
}

__global__ void k_count_deg(const int* __restrict__ dst, float* __restrict__ deg, int E) {
    int e = blockIdx.x * blockDim.x + threadIdx.x;
    if (e < E) atomicAdd(&deg[dst[e]], 1.0f);
}

__global__ void k_rsqrt(float* __restrict__ d, int n) {
    int i = blockIdx.x * blockDim.x + threadIdx.x;
    if (i < n) d[i] = rsqrtf(d[i]);
}

__global__ void k_coeff(const int* __restrict__ src, const int* __restrict__ dst,
                        const float* __restrict__ dis, float* __restrict__ coeff, int E) {
    int e = blockIdx.x * blockDim.x + threadIdx.x;
    if (e < E) coeff[e] = dis[src[e]] * dis[dst[e]];
}

// pad x[N,7] -> xp[N,8] (zero col 7) so the K=8 gemm has unconditional loads
__global__ void k_pad_x(const float* __restrict__ x, float* __restrict__ xp, int N) {
    int i = blockIdx.x * blockDim.x + threadIdx.x;
    if (i < N * 8) {
        int node = i >> 3, c = i & 7;
        xp[i] = (c < 7) ? x[node * 7 + c] : 0.0f;
    }
}

// ---------- WMMA FP32 GEMM: HW[N,64] = H[N,K] @ W[K,64], K compile-time ----------
// 256 threads = 8 waves; each wave owns one 16x16 output tile. Requires N % 16 == 0.
template <int K>
__global__ void k_gemm_wmma(const float* __restrict__ H, const float* __restrict__ W,
                            float* __restrict__ HW, int N) {
    const int wave = threadIdx.x >> 5;          // 0..7
    const int lane = threadIdx.x & 31;
    const int colTile = wave & 3;               // 0..3 -> col0 = 0..48
    const int rowTile = blockIdx.x * 2 + (wave >> 2);
    const int row0 = rowTile * 16;
    if (row0 >= N) return;                      // wave-uniform: EXEC stays all-ones

    const int mn = lane & 15;                   // A row / B,D col within tile
    const int kp = (lane >> 4) << 1;            // K sub-pair: 0 or 2
    const int col0 = colTile * 16;
    const float* __restrict__ arow = H + (size_t)(row0 + mn) * K;
    const float* __restrict__ bcol = W + col0 + mn;

    v8f acc = {};
#pragma unroll
    for (int k = 0; k < K; k += 4) {
        const int k0 = k + kp;
        // A 16x4 f32: VGPR0 = K=kp, VGPR1 = K=kp+1 (lanes 16-31 -> +2); pair is
        // contiguous & 8B-aligned -> single global_load_b64
        v2f a = *(const v2f*)(arow + k0);
        // B 4x16 f32: row-striped across lanes
        v2f b;
        b.x = bcol[(size_t)k0 * HID];
        b.y = bcol[(size_t)(k0 + 1) * HID];
        acc = __builtin_amdgcn_wmma_f32_16x16x4_f32(
            /*neg_a=*/false, a, /*neg_b=*/false, b,
            /*c_mod=*/(short)0, acc, /*reuse_a=*/false, /*reuse_b=*/false);
    }

    // D 16x16 f32: VGPR i holds M=i (lanes 0-15) / M=i+8 (lanes 16-31), N=lane&15
    const int mbase = (lane >> 4) * 8;
    float* __restrict__ dout = HW + (size_t)(row0 + mbase) * HID + col0 + mn;
#pragma unroll
    for (int i = 0; i < 8; ++i) dout[(size_t)i * HID] = acc[i];
}

// agg[i,f] = dis[i]^2 * hw[i,f] + b[f]   (self-loop term + bias as init), float4
__global__ void k_init_agg(const float4* __restrict__ hw, const float* __restrict__ dis,
                           const float4* __restrict__ b, float4* __restrict__ agg, int total4) {
    int i = blockIdx.x * blockDim.x + threadIdx.x;
    if (i < total4) {
        int node = i >> 4, f4 = i & 15;
        float d = dis[node];
        float d2 = d * d;
        float4 h = hw[i], bb = b[f4];
        agg[i] = make_float4(d2 * h.x + bb.x, d2 * h.y + bb.y,
                             d2 * h.z + bb.z, d2 * h.w + bb.w);
    }
}

// edge scatter: agg[dst,f0..f0+3] += coeff[e] * hw[src,f0..f0+3]
// thread = (edge, 4-feature chunk); lanes cover one edge's 256 B contiguously.
__global__ void k_edge_agg(const float4* __restrict__ hw, const float* __restrict__ coeff,
                           const int* __restrict__ src, const int* __restrict__ dst,
                           float* __restrict__ agg, int total16) {
    int i = blockIdx.x * blockDim.x + threadIdx.x;
    if (i < total16) {
        int e = i >> 4, c4 = i & 15;
        int s = src[e], d = dst[e];
        float w = coeff[e];
        float4 h = hw[(size_t)s * 16 + c4];
        float* out = agg + (size_t)d * HID + c4 * 4;
        atomicAdd(out + 0, w * h.x);
        atomicAdd(out + 1, w * h.y);
        atomicAdd(out + 2, w * h.z);
        atomicAdd(out + 3, w * h.w);
    }
}

__global__ void k_relu_copy(const float4* __restrict__ agg, float4* __restrict__ h,
                            int total4, int doRelu) {
    int i = blockIdx.x * blockDim.x + threadIdx.x;
    if (i < total4) {
        float4 v = agg[i];
        if (doRelu) {
            v.x = fmaxf(v.x, 0.0f); v.y = fmaxf(v.y, 0.0f);
            v.z = fmaxf(v.z, 0.0f); v.w = fmaxf(v.w, 0.0f);
        }
        h[i] = v;
    }
}

__global__ void k_pool_sum(const float4* __restrict__ h, const int* __restrict__ batch,
                           float* __restrict__ pooled, int total4) {
    int i = blockIdx.x * blockDim.x + threadIdx.x;
    if (i < total4) {
        int node = i >> 4, c4 = i & 15;
        float4 v = h[i];
        float* out = pooled + (size_t)batch[node] * HID + c4 * 4;
        atomicAdd(out + 0, v.x);
        atomicAdd(out + 1, v.y);
        atomicAdd(out + 2, v.z);
        atomicAdd(out + 3, v.w);
    }
}

__global__ void k_pool_cnt(const int* __restrict__ batch, float* __restrict__ cnt, int N) {
    int i = blockIdx.x * blockDim.x + threadIdx.x;
    if (i < N) atomicAdd(&cnt[batch[i]], 1.0f);
}

__global__ void k_head(const float* __restrict__ pooled, const float* __restrict__ cnt,
                       const float* __restrict__ lw, const float* __restrict__ lb,
                       float* __restrict__ out, int G, int O) {
    int i = blockIdx.x * blockDim.x + threadIdx.x;
    if (i < G * O) {
        int g = i / O, o = i % O;
        float s = 0.0f;
#pragma unroll
        for (int f = 0; f < HID; ++f) s += pooled[(size_t)g * HID + f] * lw[f * O + o];
        out[i] = s / fmaxf(cnt[g], 1.0f) + lb[o];
    }
}

// ---------- orchestration ----------
extern "C" void kernel_launch(void* const* d_in, const int* in_sizes, int n_in,
                              void* d_out, int out_size, void* d_ws, size_t ws_size,
                              hipStream_t stream) {
    const float* x     = (const float*)d_in[0];
    const int*   ei    = (const int*)d_in[1];
    const int*   batch = (const int*)d_in[2];
    const float* W0 = (const float*)d_in[3];
    const float* b0 = (const float*)d_in[4];
    const float* W1 = (const float*)d_in[5];
    const float* b1 = (const float*)d_in[6];
    const float* W2 = (const float*)d_in[7];
    const float* b2 = (const float*)d_in[8];
    const float* lw = (const float*)d_in[9];
    const float* lb = (const float*)d_in[10];

    const int IN_CH = 7;
    const int N = in_sizes[0] / IN_CH;        // 100000 (multiple of 16)
    const int E = in_sizes[1] / 2;            // 1200000
    const int O = in_sizes[10];               // 10
    const int G = out_size / O;               // 256
    const int* src = ei;
    const int* dst = ei + E;

    // workspace carve-up (floats; all offsets multiples of 4 -> 16B alignment)
    float* ws = (float*)d_ws;
    size_t off = 0;
    float* dis    = ws + off; off += (size_t)N;
    float* xp     = ws + off; off += (size_t)N * 8;
    float* hA     = ws + off; off += (size_t)N * HID;
    float* hB     = ws + off; off += (size_t)N * HID;
    float* hw     = ws + off; off += (size_t)N * HID;
    float* agg    = ws + off; off += (size_t)N * HID;
    float* coeff  = ws + off; off += (size_t)E;
    float* pooled = ws + off; off += (size_t)G * HID;
    float* cnt    = ws + off; off += (size_t)G;
    (void)ws_size; (void)n_in;

    const int NF4  = N * (HID / 4);           // 1.6M float4 elems
    const int EF16 = E * 16;                  // 19.2M (edge,4-feat) work items
    auto blks = [](long long n) { return (int)((n + TPB - 1) / TPB); };

    // degrees incl. self-loop, dis = rsqrt(deg), per-edge coefficient (once)
    k_fill<<<blks(N), TPB, 0, stream>>>(dis, 1.0f, N);
    k_count_deg<<<blks(E), TPB, 0, stream>>>(dst, dis, E);
    k_rsqrt<<<blks(N), TPB, 0, stream>>>(dis, N);
    k_coeff<<<blks(E), TPB, 0, stream>>>(src, dst, dis, coeff, E);
    k_pad_x<<<blks((long long)N * 8), TPB, 0, stream>>>(x, xp, N);

    const int gemmGrid = ((N + 15) / 16 + 1) / 2;   // 2 row-tiles per block

    const float* layerIn[3]  = { xp, hA, hB };
    float*       layerOut[3] = { hA, hB, hA };
    const float* Ws[3] = { W0, W1, W2 };
    const float* bs[3] = { b0, b1, b2 };

    for (int l = 0; l < 3; ++l) {
        if (l == 0)
            k_gemm_wmma<8><<<gemmGrid, TPB, 0, stream>>>(layerIn[l], Ws[l], hw, N);
        else
            k_gemm_wmma<64><<<gemmGrid, TPB, 0, stream>>>(layerIn[l], Ws[l], hw, N);
        k_init_agg<<<blks(NF4), TPB, 0, stream>>>((const float4*)hw, dis,
                                                  (const float4*)bs[l], (float4*)agg, NF4);
        k_edge_agg<<<blks(EF16), TPB, 0, stream>>>((const float4*)hw, coeff, src, dst,
                                                   agg, EF16);
        k_relu_copy<<<blks(NF4), TPB, 0, stream>>>((const float4*)agg, (float4*)layerOut[l],
                                                   NF4, l < 2 ? 1 : 0);
    }

    // global mean pool + linear head (pooled & cnt contiguous -> one zero-fill)
    k_fill<<<blks((long long)G * HID + G), TPB, 0, stream>>>(pooled, 0.0f, G * HID + G);
    k_pool_sum<<<blks(NF4), TPB, 0, stream>>>((const float4*)hA, batch, pooled, NF4);
    k_pool_cnt<<<blks(N), TPB, 0, stream>>>(batch, cnt, N);
    k_head<<<blks((long long)G * O), TPB, 0, stream>>>(pooled, cnt, lw, lb, (float*)d_out, G, O);
}